// RangeLoss_24541443130053
// MI455X (gfx1250) — compile-verified
//
#include <hip/hip_runtime.h>
#include <math.h>

typedef float v2f __attribute__((ext_vector_type(2)));
typedef float v8f __attribute__((ext_vector_type(8)));

#define BATCH 8192
#define FEAT  256
#define NCLS  1000
#define TILE  128
#define KC    32
#define LDSS  36   // padded LDS row stride in floats (bank spread + 16B-aligned float4 stores)

// ---------------------------------------------------------------- init scratch
__global__ void init_ws_kernel(float* ws) {
    ws[0] = 0.0f;      // running max of masked relu(d^2)
    ws[1] = 3.0e38f;   // running min of clipped inter distances
}

// ---------------------------------------------------------------- row |x_i|^2
__global__ __launch_bounds__(256)
void row_sumsq_kernel(const float* __restrict__ x, float* __restrict__ mag) {
    int gid  = blockIdx.x * blockDim.x + threadIdx.x;
    int wid  = gid >> 5;          // one wave32 per row
    int lane = gid & 31;
    if (wid >= BATCH) return;
    const float4* p = (const float4*)(x + (size_t)wid * FEAT);
    float4 a = p[lane * 2 + 0];
    float4 b = p[lane * 2 + 1];
    float s = a.x*a.x + a.y*a.y + a.z*a.z + a.w*a.w
            + b.x*b.x + b.y*b.y + b.z*b.z + b.w*b.w;
    #pragma unroll
    for (int off = 16; off > 0; off >>= 1) s += __shfl_xor(s, off);
    if (lane == 0) mag[wid] = s;
}

// ------------------------------------------- WMMA tile kernel: max masked d^2
// Upper-triangular 128x128 tiles of sim = x @ x^T via V_WMMA_F32_16X16X4_F32.
// 8 waves/block in a 2x4 layout; each wave owns a 64x32 sub-tile (4x2 accums).
__global__ __launch_bounds__(256)
void intra_max_kernel(const float* __restrict__ x,
                      const int*   __restrict__ labels,
                      const float* __restrict__ mag,
                      float* ws) {
    const int bi = blockIdx.x;
    const int bj = blockIdx.y;
    if (bj < bi) return;   // symmetry: upper triangle only (uniform across block)

    __shared__ float As[TILE * LDSS];
    __shared__ float Bs[TILE * LDSS];
    __shared__ float red[8];

    const int tid   = threadIdx.x;
    const int lane  = tid & 31;
    const int wave  = tid >> 5;
    const int waveM = wave >> 2;            // 0..1  (64-row slab)
    const int waveN = wave & 3;             // 0..3  (32-col slab)
    const int rowA0 = bi * TILE;
    const int rowB0 = bj * TILE;
    const int l15   = lane & 15;
    const int kbl   = (lane >> 4) << 1;     // K-pair select: lanes 0-15 -> {K0,K1}, 16-31 -> {K2,K3}

    v8f acc[4][2];
    #pragma unroll
    for (int mt = 0; mt < 4; ++mt)
        #pragma unroll
        for (int nt = 0; nt < 2; ++nt)
            acc[mt][nt] = {};

    for (int kc = 0; kc < FEAT / KC; ++kc) {
        // stage 128x32 slabs of both row-blocks into LDS (coalesced float4)
        #pragma unroll
        for (int p = 0; p < 4; ++p) {
            int idx = p * 256 + tid;        // 0..1023
            int r   = idx >> 3;             // 0..127
            int c4  = (idx & 7) << 2;       // 0,4,...,28
            *(float4*)&As[r * LDSS + c4] =
                *(const float4*)&x[(size_t)(rowA0 + r) * FEAT + kc * KC + c4];
            *(float4*)&Bs[r * LDSS + c4] =
                *(const float4*)&x[(size_t)(rowB0 + r) * FEAT + kc * KC + c4];
        }
        __syncthreads();

        #pragma unroll
        for (int kk = 0; kk < KC / 4; ++kk) {
            const int kb = kk * 4 + kbl;
            v2f a[4], b[2];
            #pragma unroll
            for (int mt = 0; mt < 4; ++mt)
                a[mt] = *(const v2f*)&As[(waveM * 64 + mt * 16 + l15) * LDSS + kb];
            #pragma unroll
            for (int nt = 0; nt < 2; ++nt)
                b[nt] = *(const v2f*)&Bs[(waveN * 32 + nt * 16 + l15) * LDSS + kb];
            #pragma unroll
            for (int mt = 0; mt < 4; ++mt)
                #pragma unroll
                for (int nt = 0; nt < 2; ++nt)
                    acc[mt][nt] = __builtin_amdgcn_wmma_f32_16x16x4_f32(
                        false, a[mt], false, b[nt], (short)0, acc[mt][nt], false, false);
        }
        __syncthreads();
    }

    // epilogue: d2 = |xi|^2 + |xj|^2 - 2 sim; mask by labels; max-reduce
    // C/D layout: VGPR r -> M = r + 8*(lane>=16), N = lane&15
    float lmax = 0.0f;
    const int lhi = (lane >> 4) << 3;
    float magj[2]; int labj[2];
    #pragma unroll
    for (int nt = 0; nt < 2; ++nt) {
        int j   = rowB0 + waveN * 32 + nt * 16 + l15;
        magj[nt] = mag[j];
        labj[nt] = labels[j];
    }
    #pragma unroll
    for (int mt = 0; mt < 4; ++mt) {
        #pragma unroll
        for (int r = 0; r < 8; ++r) {
            int i = rowA0 + waveM * 64 + mt * 16 + lhi + r;
            float magi = mag[i];
            int   labi = labels[i];
            #pragma unroll
            for (int nt = 0; nt < 2; ++nt) {
                float d2 = magi + magj[nt] - 2.0f * acc[mt][nt][r];
                if (labi == labj[nt]) lmax = fmaxf(lmax, fmaxf(d2, 0.0f));
            }
        }
    }
    #pragma unroll
    for (int off = 16; off > 0; off >>= 1) lmax = fmaxf(lmax, __shfl_xor(lmax, off));
    if (lane == 0) red[wave] = lmax;
    __syncthreads();
    if (tid == 0) {
        float m = red[0];
        #pragma unroll
        for (int w = 1; w < 8; ++w) m = fmaxf(m, red[w]);
        // values are >= 0 so IEEE bit pattern is monotone under int compare
        atomicMax((int*)&ws[0], __float_as_int(m));
    }
}

// -------------------------------------------------- inter: min clipped 2*cj.(cj-ci)
__global__ __launch_bounds__(256)
void inter_min_kernel(const float* __restrict__ centers, float* ws) {
    __shared__ float red[8];
    int t = blockIdx.x * blockDim.x + threadIdx.x;
    float v = 3.0e38f;
    if (t < NCLS * NCLS) {
        int i = t / NCLS;
        int j = t - i * NCLS;
        const float4* ci = (const float4*)(centers + (size_t)i * FEAT);
        const float4* cj = (const float4*)(centers + (size_t)j * FEAT);
        float s = 0.0f;
        #pragma unroll 4
        for (int k = 0; k < FEAT / 4; ++k) {
            float4 a = ci[k], b = cj[k];
            s += b.x * (b.x - a.x) + b.y * (b.y - a.y)
               + b.z * (b.z - a.z) + b.w * (b.w - a.w);
        }
        v = 2.0f * s;
        v = fminf(fmaxf(v, 1e-12f), 1e12f);
    }
    #pragma unroll
    for (int off = 16; off > 0; off >>= 1) v = fminf(v, __shfl_xor(v, off));
    int lane = threadIdx.x & 31, wave = threadIdx.x >> 5;
    if (lane == 0) red[wave] = v;
    __syncthreads();
    if (threadIdx.x == 0) {
        float m = red[0];
        #pragma unroll
        for (int w = 1; w < 8; ++w) m = fminf(m, red[w]);
        atomicMin((int*)&ws[1], __float_as_int(m));  // all values > 0
    }
}

// ---------------------------------------------------------------- final scalar
__global__ void finalize_kernel(const float* ws, float* out) {
    // top-2 of the symmetric masked matrix are both d_max -> harmonic term = d_max
    float d = sqrtf(fmaxf(ws[0], 0.0f));
    d = fminf(fmaxf(d, 1e-12f), 1e12f);
    float loss_intra = d;
    float loss_inter = fminf(fmaxf(5.0f - ws[1], 0.0f), 1.0e6f);
    out[0] = 1.0f * loss_intra + 1.0f * loss_inter;   // ALPHA=BETA=1
}

extern "C" void kernel_launch(void* const* d_in, const int* in_sizes, int n_in,
                              void* d_out, int out_size, void* d_ws, size_t ws_size,
                              hipStream_t stream) {
    (void)in_sizes; (void)n_in; (void)out_size; (void)ws_size;
    const float* x       = (const float*)d_in[0];
    const int*   labels  = (const int*)d_in[1];
    const float* centers = (const float*)d_in[2];
    float* out = (float*)d_out;
    float* ws  = (float*)d_ws;
    float* mag = ws + 2;                  // 8192 floats of scratch after the two scalars

    init_ws_kernel<<<1, 1, 0, stream>>>(ws);
    row_sumsq_kernel<<<(BATCH * 32) / 256, 256, 0, stream>>>(x, mag);

    dim3 grid(BATCH / TILE, BATCH / TILE);
    intra_max_kernel<<<grid, 256, 0, stream>>>(x, labels, mag, ws);

    int nInter = NCLS * NCLS;
    inter_min_kernel<<<(nInter + 255) / 256, 256, 0, stream>>>(centers, ws);

    finalize_kernel<<<1, 1, 0, stream>>>(ws, out);
}